// Glm4vCubingModule_82755429859780
// MI455X (gfx1250) — compile-verified
//
#include <hip/hip_runtime.h>
#include <hip/hip_bf16.h>
#include <math.h>

#define N_FRAMES 128
#define P_PATCH  576
#define D_DIM    1536
#define LM_DIM   4096
#define NQ_DIM   64
#define NUM_CUBES 8
#define ALPHA_C 0.5f
#define TEMP_C  0.5f
#define LR_G_C  0.1f
#define G_EPS_C 1e-20f

typedef float v2f __attribute__((ext_vector_type(2)));
typedef float v8f __attribute__((ext_vector_type(8)));
typedef int   v4i_vs __attribute__((vector_size(16)));   // matches builtin param type

#define AS_GLOBAL __attribute__((address_space(1)))
#define AS_LDS    __attribute__((address_space(3)))

#if defined(__has_builtin)
#if __has_builtin(__builtin_amdgcn_global_load_async_to_lds_b128) && \
    __has_builtin(__builtin_amdgcn_s_wait_asynccnt)
#define USE_ASYNC_LDS 1
#endif
#endif
#ifndef USE_ASYNC_LDS
#define USE_ASYNC_LDS 0
#endif

#if USE_ASYNC_LDS
// flat->AS1 via integer round-trip (identity for global); flat low 32 bits are
// the LDS byte offset.
__device__ __forceinline__ void async_cp16(const float* g, float* lds_flat) {
  __builtin_amdgcn_global_load_async_to_lds_b128(
      (AS_GLOBAL v4i_vs*)(uintptr_t)g,
      (AS_LDS v4i_vs*)(unsigned int)(uintptr_t)lds_flat,
      0, 0);
}
#endif

// ---------------------------------------------------------------------------
// K1: per-frame patch partial sums over 1/4 of the patches (the 452MB pass).
// grid (128 frames, 4 chunks), block 384 threads (float4 across D=1536).
// ---------------------------------------------------------------------------
__global__ void k_patch_partial_sum(const float* __restrict__ vf, float* __restrict__ part) {
  int n = blockIdx.x;
  int c = blockIdx.y;            // patch chunk, 144 patches each
  int t = threadIdx.x;           // 384 threads * float4 = 1536 columns
  const float4* base = (const float4*)(vf + ((size_t)n * P_PATCH + (size_t)c * 144) * D_DIM) + t;
  float4 acc = make_float4(0.f, 0.f, 0.f, 0.f);
  for (int p = 0; p < 144; ++p) {
    if (p + 8 < 144)
      __builtin_prefetch((const void*)(base + (size_t)(p + 8) * (D_DIM / 4)), 0, 0);
    float4 v = base[(size_t)p * (D_DIM / 4)];
    acc.x += v.x; acc.y += v.y; acc.z += v.z; acc.w += v.w;
  }
  ((float4*)(part + ((size_t)c * N_FRAMES + n) * D_DIM))[t] = acc;
}

// ---------------------------------------------------------------------------
// K2: combine partials -> per-frame patch mean, frame-diff, EMA (serial in n,
// parallel in d). Writes mom_mean[127x1536] + zeroed pad row 127.
// ---------------------------------------------------------------------------
__global__ void k_mom_mean(const float* __restrict__ part, float* __restrict__ mom_mean) {
  int d = blockIdx.x * blockDim.x + threadIdx.x;
  if (d >= D_DIM) return;
  const float invP = 1.0f / (float)P_PATCH;
  float prev;
  {
    float s = 0.f;
    #pragma unroll
    for (int c = 0; c < 4; ++c) s += part[((size_t)c * N_FRAMES + 0) * D_DIM + d];
    prev = s * invP;
  }
  float mom = 0.f;
  for (int n = 0; n < N_FRAMES - 1; ++n) {
    float s = 0.f;
    #pragma unroll
    for (int c = 0; c < 4; ++c) s += part[((size_t)c * N_FRAMES + (n + 1)) * D_DIM + d];
    float cur = s * invP;
    float diff = cur - prev;
    prev = cur;
    mom = (n == 0) ? diff : (ALPHA_C * diff + (1.0f - ALPHA_C) * mom);
    mom_mean[(size_t)n * D_DIM + d] = mom;
  }
  mom_mean[(size_t)(N_FRAMES - 1) * D_DIM + d] = 0.0f;   // pad row
}

// ---------------------------------------------------------------------------
// K3/K5/K9: f32 WMMA GEMM, double-buffered async-LDS staged.
// C[MxN] = A[MxK] @ B[KxN] + bias, optional exact GELU.
// Block = 128 threads (4 waves) -> 16(M) x 64(N) tile.  K in KC=64 chunks:
// while chunk i is computed (16 x V_WMMA_F32_16X16X4_F32 from LDS), chunk
// i+1's 10 GLOBAL_LOAD_ASYNC_TO_LDS_B128 per thread are already in flight;
// s_wait_asynccnt(10) retires the older chunk (async loads complete in
// order), barrier publishes it block-wide.
// LDS strides: As=68 (A reads bank-conflict-free: bank = 4*l16 + c),
//              Bs=72 (half-waves hit disjoint 16-bank windows). 16B aligned.
// WMMA layouts (ISA 7.12.2, 32-bit):
//   A(16x4): lane l -> row l%16, K = vgpr + 2*(l>=16)
//   B(4x16): lane l -> col l%16, K = vgpr + 2*(l>=16)
//   C(16x16): vgpr r -> M = r + 8*(l>=16), N = l%16
// ---------------------------------------------------------------------------
#define KC 64
#define AS_STRIDE 68
#define BS_STRIDE 72

__global__ __launch_bounds__(128) void k_wmma_gemm(
    const float* __restrict__ A, const float* __restrict__ B,
    const float* __restrict__ bias, float* __restrict__ C,
    int K, int N, int M_store, int apply_gelu) {
  __shared__ float As[2][16 * AS_STRIDE];
  __shared__ float Bs[2][KC * BS_STRIDE];

  int tid  = threadIdx.x;
  int wave = tid >> 5;
  int lane = tid & 31;
  int half = lane >> 4;
  int l16  = lane & 15;
  int nb   = blockIdx.x * 64;
  int n    = nb + wave * 16 + l16;
  int mt   = blockIdx.y * 16;

  v8f c;
  float bv = bias[n];
  #pragma unroll
  for (int r = 0; r < 8; ++r) c[r] = bv;

  // stage one KC-chunk (10 x 16B per thread) into buffer `buf`
  auto stage = [&](int k0, int buf) {
    #pragma unroll
    for (int p = 0; p < 2; ++p) {          // A: 16x64 = 1024 floats
      int idx = p * 512 + tid * 4;
      int r  = idx >> 6;
      int kk = idx & 63;
      const float* g = A + (size_t)(mt + r) * K + k0 + kk;
      float* l = &As[buf][r * AS_STRIDE + kk];
#if USE_ASYNC_LDS
      async_cp16(g, l);
#else
      *(float4*)l = *(const float4*)g;
#endif
    }
    #pragma unroll
    for (int p = 0; p < 8; ++p) {          // B: 64x64 = 4096 floats
      int idx = p * 512 + tid * 4;
      int kk = idx >> 6;
      int nn = idx & 63;
      const float* g = B + (size_t)(k0 + kk) * N + nb + nn;
      float* l = &Bs[buf][kk * BS_STRIDE + nn];
#if USE_ASYNC_LDS
      async_cp16(g, l);
#else
      *(float4*)l = *(const float4*)g;
#endif
    }
  };

  int nchunks = K / KC;
  stage(0, 0);
  for (int i = 0; i < nchunks; ++i) {
    if (i + 1 < nchunks) {
      // buffer (i+1)&1 was last read in chunk i-1; the barrier ending chunk
      // i-1 makes this issue safe for all waves.
      stage((i + 1) * KC, (i + 1) & 1);
#if USE_ASYNC_LDS
      __builtin_amdgcn_s_wait_asynccnt(10);   // retire chunk i's copies only
#endif
    } else {
#if USE_ASYNC_LDS
      __builtin_amdgcn_s_wait_asynccnt(0);
#endif
    }
    __syncthreads();                          // chunk i visible block-wide

    const float* Asb = As[i & 1];
    const float* Bsb = Bs[i & 1];
    #pragma unroll
    for (int kk = 0; kk < KC; kk += 4) {
      v2f a, b;
      int ak = kk + 2 * half;
      a.x = Asb[l16 * AS_STRIDE + ak];
      a.y = Asb[l16 * AS_STRIDE + ak + 1];
      b.x = Bsb[ak * BS_STRIDE + wave * 16 + l16];
      b.y = Bsb[(ak + 1) * BS_STRIDE + wave * 16 + l16];
      c = __builtin_amdgcn_wmma_f32_16x16x4_f32(false, a, false, b, (short)0, c, false, false);
    }
    __syncthreads();                          // all waves done reading buf i&1
  }

  int m_base = mt + 8 * half;
  #pragma unroll
  for (int r = 0; r < 8; ++r) {
    float v = c[r];
    if (apply_gelu) v = 0.5f * v * (1.0f + erff(v * 0.70710678118654752f)); // exact GELU
    int m = m_base + r;
    if (m < M_store) C[(size_t)m * N + n] = v;
  }
}

// ---------------------------------------------------------------------------
// K4: LayerNorm per row (127 rows) with gamma/beta; zero pad row 127.
// ---------------------------------------------------------------------------
__global__ void k_layernorm(const float* __restrict__ feat, const float* __restrict__ g,
                            const float* __restrict__ b, float* __restrict__ xn) {
  int row = blockIdx.x;
  int t = threadIdx.x;  // 256
  if (row == N_FRAMES - 1) {
    for (int d = t; d < D_DIM; d += 256) xn[(size_t)row * D_DIM + d] = 0.0f;
    return;
  }
  __shared__ float red[256];
  const float* x = feat + (size_t)row * D_DIM;
  float s = 0.f;
  for (int d = t; d < D_DIM; d += 256) s += x[d];
  red[t] = s; __syncthreads();
  for (int o = 128; o > 0; o >>= 1) { if (t < o) red[t] += red[t + o]; __syncthreads(); }
  float mu = red[0] * (1.0f / (float)D_DIM);
  __syncthreads();
  float v = 0.f;
  for (int d = t; d < D_DIM; d += 256) { float e = x[d] - mu; v += e * e; }
  red[t] = v; __syncthreads();
  for (int o = 128; o > 0; o >>= 1) { if (t < o) red[t] += red[t + o]; __syncthreads(); }
  float rstd = rsqrtf(red[0] * (1.0f / (float)D_DIM) + 1e-5f);
  for (int d = t; d < D_DIM; d += 256)
    xn[(size_t)row * D_DIM + d] = (x[d] - mu) * rstd * g[d] + b[d];
}

// ---------------------------------------------------------------------------
// K6: gate logits (h @ W2[1536x2] + b2), Gumbel softmax -> y[127].
// ---------------------------------------------------------------------------
__global__ void k_gate(const float* __restrict__ h, const float* __restrict__ W2,
                       const float* __restrict__ b2, const float* __restrict__ gu,
                       float* __restrict__ out_logits, float* __restrict__ y) {
  int row = blockIdx.x;   // 0..126
  int t = threadIdx.x;    // 256
  __shared__ float r0[256], r1[256];
  const float* x = h + (size_t)row * D_DIM;
  float s0 = 0.f, s1 = 0.f;
  for (int d = t; d < D_DIM; d += 256) {
    float xv = x[d];
    s0 += xv * W2[2 * d + 0];
    s1 += xv * W2[2 * d + 1];
  }
  r0[t] = s0; r1[t] = s1; __syncthreads();
  for (int o = 128; o > 0; o >>= 1) {
    if (t < o) { r0[t] += r0[t + o]; r1[t] += r1[t + o]; }
    __syncthreads();
  }
  if (t == 0) {
    float l0 = r0[0] + b2[0];
    float l1 = r1[0] + b2[1];
    out_logits[2 * row + 0] = l0;
    out_logits[2 * row + 1] = l1;
    float g0 = -logf(-logf(gu[2 * row + 0] + G_EPS_C) + G_EPS_C);
    float g1 = -logf(-logf(gu[2 * row + 1] + G_EPS_C) + G_EPS_C);
    float a0 = (l0 + g0 * LR_G_C) / TEMP_C;
    float a1 = (l1 + g1 * LR_G_C) / TEMP_C;
    float m  = fmaxf(a0, a1);
    float e0 = expf(a0 - m), e1 = expf(a1 - m);
    y[row] = e1 / (e0 + e1);
  }
}

// ---------------------------------------------------------------------------
// K7: top-8 of y (strict > scan: lower index wins ties, matching top_k),
// build z_hard[128] (selected: (1-y)+y, else exact 0, z[0]=1) and zsum.
// ---------------------------------------------------------------------------
__global__ void k_topk_z(const float* __restrict__ y, float* __restrict__ zbuf,
                         float* __restrict__ out_z) {
  __shared__ float zs[N_FRAMES + 1];
  int t = threadIdx.x;   // 128
  if (t == 0) {
    bool sel[N_FRAMES - 1];
    for (int i = 0; i < N_FRAMES - 1; ++i) sel[i] = false;
    for (int k = 0; k < NUM_CUBES; ++k) {
      float best = -INFINITY; int bi = -1;
      for (int i = 0; i < N_FRAMES - 1; ++i)
        if (!sel[i] && y[i] > best) { best = y[i]; bi = i; }
      if (bi >= 0) sel[bi] = true;
    }
    zs[0] = 1.0f;
    float zsum = 1.0f;
    for (int i = 0; i < N_FRAMES - 1; ++i) {
      float yv = y[i];
      float zv = sel[i] ? ((1.0f - yv) + yv) : 0.0f;
      zs[i + 1] = zv;
      zsum += zv;
    }
    zs[N_FRAMES] = zsum;
  }
  __syncthreads();
  zbuf[t] = zs[t];
  out_z[t] = zs[t];
  if (t == 0) zbuf[N_FRAMES] = zs[N_FRAMES];
}

// ---------------------------------------------------------------------------
// K8: pooled[64x1536] = sum_n z[n] * sum_{j<9} vf[n, 9q+j, :] / (9 * zsum).
// Only ~9 frames have z != 0 -> ~32 MB of reads.
// ---------------------------------------------------------------------------
__global__ void k_pooled(const float* __restrict__ vf, const float* __restrict__ zbuf,
                         float* __restrict__ pooled) {
  int q = blockIdx.x;    // 64
  int t = threadIdx.x;   // 384 * float4 = 1536 columns
  __shared__ float zsh[N_FRAMES + 1];
  for (int i = t; i < N_FRAMES + 1; i += blockDim.x) zsh[i] = zbuf[i];
  __syncthreads();
  float inv = 1.0f / (9.0f * zsh[N_FRAMES]);
  float4 acc = make_float4(0.f, 0.f, 0.f, 0.f);
  for (int n = 0; n < N_FRAMES; ++n) {
    float zn = zsh[n];
    if (zn != 0.0f) {
      const float4* base = (const float4*)(vf + ((size_t)n * P_PATCH + (size_t)q * 9) * D_DIM) + t;
      #pragma unroll
      for (int j = 0; j < 9; ++j) {
        float4 v = base[(size_t)j * (D_DIM / 4)];
        acc.x += zn * v.x; acc.y += zn * v.y; acc.z += zn * v.z; acc.w += zn * v.w;
      }
    }
  }
  acc.x *= inv; acc.y *= inv; acc.z *= inv; acc.w *= inv;
  ((float4*)(pooled + (size_t)q * D_DIM))[t] = acc;
}

// ---------------------------------------------------------------------------
extern "C" void kernel_launch(void* const* d_in, const int* in_sizes, int n_in,
                              void* d_out, int out_size, void* d_ws, size_t ws_size,
                              hipStream_t stream) {
  const float* vf    = (const float*)d_in[0];
  const float* gu    = (const float*)d_in[1];
  const float* W_agg = (const float*)d_in[2];
  const float* b_agg = (const float*)d_in[3];
  const float* ln_g  = (const float*)d_in[4];
  const float* ln_b  = (const float*)d_in[5];
  const float* W1    = (const float*)d_in[6];
  const float* b1    = (const float*)d_in[7];
  const float* W2    = (const float*)d_in[8];
  const float* b2    = (const float*)d_in[9];
  const float* W_th  = (const float*)d_in[10];
  const float* b_th  = (const float*)d_in[11];

  float* out        = (float*)d_out;
  float* out_logits = out;                                   // 127*2 = 254
  float* out_thumb  = out + 254;                             // 64*4096 = 262144
  float* out_z      = out + 254 + (size_t)NQ_DIM * LM_DIM;   // 128

  float* ws = (float*)d_ws;
  size_t off = 0;
  float* part     = ws + off; off += (size_t)4 * N_FRAMES * D_DIM;   // 786432
  float* mom_mean = ws + off; off += (size_t)N_FRAMES * D_DIM;       // 196608
  float* feat     = ws + off; off += (size_t)N_FRAMES * D_DIM;
  float* xn       = ws + off; off += (size_t)N_FRAMES * D_DIM;
  float* hbuf     = ws + off; off += (size_t)N_FRAMES * D_DIM;
  float* ybuf     = ws + off; off += N_FRAMES;
  float* zbuf     = ws + off; off += N_FRAMES + 1;
  float* pooled   = ws + off; off += (size_t)NQ_DIM * D_DIM;

  // 1) 452 MB streaming pass: per-frame patch partial sums
  k_patch_partial_sum<<<dim3(N_FRAMES, 4), 384, 0, stream>>>(vf, part);
  // 2) patch mean -> frame diffs -> EMA -> mom_mean[127(+pad)x1536]
  k_mom_mean<<<(D_DIM + 255) / 256, 256, 0, stream>>>(part, mom_mean);
  // 3) feat = mom_mean @ W_agg + b_agg   (f32 WMMA, double-buffered async LDS)
  k_wmma_gemm<<<dim3(D_DIM / 64, N_FRAMES / 16), 128, 0, stream>>>(
      mom_mean, W_agg, b_agg, feat, D_DIM, D_DIM, N_FRAMES, 0);
  // 4) LayerNorm
  k_layernorm<<<N_FRAMES, 256, 0, stream>>>(feat, ln_g, ln_b, xn);
  // 5) h = gelu(xn @ W1 + b1)            (f32 WMMA + exact erf GELU)
  k_wmma_gemm<<<dim3(D_DIM / 64, N_FRAMES / 16), 128, 0, stream>>>(
      xn, W1, b1, hbuf, D_DIM, D_DIM, N_FRAMES, 1);
  // 6) gate logits + Gumbel-softmax y
  k_gate<<<N_FRAMES - 1, 256, 0, stream>>>(hbuf, W2, b2, gu, out_logits, ybuf);
  // 7) top-8 -> z_hard
  k_topk_z<<<1, N_FRAMES, 0, stream>>>(ybuf, zbuf, out_z);
  // 8) keyframe-weighted pooled features (reads only selected frames)
  k_pooled<<<NQ_DIM, 384, 0, stream>>>(vf, zbuf, pooled);
  // 9) thumbnail = pooled @ W_th + b_th  (f32 WMMA)
  k_wmma_gemm<<<dim3(LM_DIM / 64, NQ_DIM / 16), 128, 0, stream>>>(
      pooled, W_th, b_th, out_thumb, D_DIM, LM_DIM, NQ_DIM, 0);
}